// BiLSTM_CRF_46978352284447
// MI455X (gfx1250) — compile-verified
//
#include <hip/hip_runtime.h>

// CRF loss on MI455X (gfx1250, wave32).
#define BB 512
#define TT 2048
#define S  24
#define START_TAG 22
#define STOP_TAG  23
#define NEGV (-10000.0f)

#define LN2f     0.6931471805599453f
#define INV_LN2f 1.4426950408889634f

// feats double-buffer chunking for TDM prefetch
#define CHUNK_T 16
#define NCH (TT / CHUNK_T)

typedef float v2f __attribute__((ext_vector_type(2)));
typedef float v8f __attribute__((ext_vector_type(8)));
typedef unsigned int u32x4 __attribute__((ext_vector_type(4)));
typedef int i32x4 __attribute__((ext_vector_type(4)));
typedef int i32x8 __attribute__((ext_vector_type(8)));

// LDS row stride for the 16x24 g tile: 32 -> columns 24..31 are in-row padding,
// so dead writeback lanes can be redirected without branching.
#define FV_STRIDE 32

#if defined(__has_builtin)
#if __has_builtin(__builtin_amdgcn_tensor_load_to_lds) && __has_builtin(__builtin_amdgcn_s_wait_tensorcnt)
#define USE_TDM 1
#endif
#endif
#ifndef USE_TDM
#define USE_TDM 0
#endif

// Raw hardware transcendentals: v_exp_f32 is exp2, v_log_f32 is log2.
__device__ __forceinline__ float fast_exp2(float x) {
#if defined(__has_builtin) && __has_builtin(__builtin_amdgcn_exp2f)
  return __builtin_amdgcn_exp2f(x);
#else
  return __expf(x * LN2f);
#endif
}
__device__ __forceinline__ float fast_log2(float x) {
#if defined(__has_builtin) && __has_builtin(__builtin_amdgcn_logf)
  return __builtin_amdgcn_logf(x);
#else
  return __logf(x) * INV_LN2f;
#endif
}
// xor-16 half swap without LDS (v_permlanex16_b32, identity lane selects).
__device__ __forceinline__ float halfswap16(float v) {
#if defined(__has_builtin) && __has_builtin(__builtin_amdgcn_permlanex16)
  const int s = __float_as_int(v);
  return __int_as_float(
      __builtin_amdgcn_permlanex16(s, s, 0x76543210, 0xfedcba98, false, false));
#else
  return __shfl_xor(v, 16, 32);
#endif
}

#if USE_TDM
// TDM load of one feats chunk: 16 rows (batch) x (CHUNK_T*S) contiguous f32
// per row, row stride TT*S elements, into LDS at byte address `ldsb`.
// D# per CDNA5 ISA (2D tile; groups 2/3/4 unused).
__device__ __forceinline__ void tdm_feats_load(const float* gsrc, unsigned ldsb) {
  const unsigned long long ga = (unsigned long long)(uintptr_t)gsrc;
  u32x4 g0;
  g0.x = 1u;                                   // count=1, user descriptor
  g0.y = ldsb;                                 // lds_addr (bytes)
  g0.z = (unsigned)ga;                         // global_addr[31:0]
  g0.w = ((unsigned)(ga >> 32) & 0x01FFFFFFu)  // global_addr[56:32]
         | 0x80000000u;                        // type=2 ("image")
  i32x8 g1;
  g1[0] = 0x00020000;                          // wg_mask=0, data_size=2 (4B)
  g1[1] = (int)(((CHUNK_T * S) & 0xFFFF) << 16);   // tensor_dim0[15:0]
  g1[2] = (int)(16 << 16);                         // dim0 hi=0 | tensor_dim1=16
  g1[3] = (int)((CHUNK_T * S) << 16);              // dim1 hi=0 | tile_dim0=384
  g1[4] = 16;                                      // tile_dim1=16, tile_dim2=0
  g1[5] = TT * S;                                  // tensor_dim0_stride[31:0]
  g1[6] = 0;
  g1[7] = 0;
  const i32x4 gz4 = {0, 0, 0, 0};
  const i32x8 gz8 = {0, 0, 0, 0, 0, 0, 0, 0};
  __builtin_amdgcn_tensor_load_to_lds(g0, g1, gz4, gz4, gz8, 0);
}
#endif

// ---------------------------------------------------------------------------
// Forward algorithm, one wave per 16 batch rows, entirely in log2 domain:
//   fv = (g + c_b) * ln2 ; the per-row offset c cancels inside
//   exp2(g - rowmax(g)), so no cross-lane broadcast is needed per step.
//   P  = exp2(g - rowmax(g))               (raw v_exp_f32, WMMA A layout)
//   D  = P (16xK) x E (KxN)                12 chained V_WMMA_F32_16X16X4_F32
//   g' = log2(D) + feat/ln2 ; c += rowmax  (frozen rows redirected to dump)
// E[k][n] = exp(trans[n][k]) = 2^(trans/ln2) in registers. feats stream
// through LDS via double-buffered TENSOR_LOAD_TO_LDS (TENSORcnt-gated).
// ---------------------------------------------------------------------------
__global__ __launch_bounds__(32) void crf_forward_wmma(
    const float* __restrict__ feats,     // [B, T, S]
    const float* __restrict__ trans,     // [S, S]
    const int*   __restrict__ lengths,   // [B]
    float*       __restrict__ alpha_out) // [B]
{
  __shared__ float g2d[16 * FV_STRIDE];
  __shared__ float dump[32];             // sink for dead writeback lanes
#if USE_TDM
  __shared__ float fstage[2][16 * CHUNK_T * S + 8];  // +8: OOB-read padding
#endif

  const int lane = threadIdx.x;       // 0..31
  const int b0   = blockIdx.x * 16;
  const int half = (lane >> 4) & 1;   // 0: lanes 0-15, 1: lanes 16-31
  const int l16  = lane & 15;

  // ---- exp-space transition matrix as WMMA B operands (registers) ----
  v2f bA[6];  // N = 0..15
  v2f bB[6];  // N = 16..23 (cols 24..31 zero)
#pragma unroll
  for (int kk = 0; kk < 6; ++kk) {
    const int k0 = 4 * kk + 2 * half;
    {
      const int n = l16;
      bA[kk].x = __expf(trans[n * S + k0]);
      bA[kk].y = __expf(trans[n * S + k0 + 1]);
    }
    {
      const int  n  = 16 + (l16 & 7);
      const bool ok = (l16 < 8);
      bB[kk].x = ok ? __expf(trans[n * S + k0]) : 0.0f;
      bB[kk].y = ok ? __expf(trans[n * S + k0 + 1]) : 0.0f;
    }
  }

  const int lenL = lengths[b0 + l16];  // length of row (lane & 15)
  int len_r[8];                        // hoisted per-writeback-row lengths
#pragma unroll
  for (int r = 0; r < 8; ++r) len_r[r] = __shfl(lenL, r + 8 * half, 32);
  // Steps beyond max(len) over the tile are no-ops: clamp the scan.
  int lenM = lenL;
#pragma unroll
  for (int off = 8; off > 0; off >>= 1)
    lenM = max(lenM, __shfl_xor(lenM, off, 32));
  const int nchEff = (lenM + CHUNK_T - 1) / CHUNK_T;

  // ---- Init g (log2 units): g[b][i] = (i==START) ? 0 : NEG/ln2 ----
  for (int idx = lane; idx < 16 * FV_STRIDE; idx += 32) {
    const int i = idx % FV_STRIDE;
    g2d[idx] = (i == START_TAG) ? 0.0f : NEGV * INV_LN2f;
  }
  float c = 0.0f;
  __syncthreads();

  const float* gchunk = feats + (size_t)b0 * TT * S;

#if USE_TDM
  const unsigned ldsA[2] = { (unsigned)(uintptr_t)&fstage[0][0],
                             (unsigned)(uintptr_t)&fstage[1][0] };
  tdm_feats_load(gchunk, ldsA[0]);   // prologue: chunk 0
  int buf = 0;
#endif

  for (int ch = 0; ch < nchEff; ++ch) {
#if USE_TDM
    if (ch + 1 < nchEff) {
      tdm_feats_load(gchunk + (size_t)(ch + 1) * CHUNK_T * S, ldsA[buf ^ 1]);
      __builtin_amdgcn_s_wait_tensorcnt(1);   // chunk `ch` resident
    } else {
      __builtin_amdgcn_s_wait_tensorcnt(0);
    }
    asm volatile("" ::: "memory");
    const float* fb = &fstage[buf][0];
#endif

    for (int tt = 0; tt < CHUNK_T; ++tt) {
      const int t = ch * CHUNK_T + tt;

      // ---- Load g from LDS in WMMA A layout (16x4 f32 chunks). ----
      v2f a[6];
#pragma unroll
      for (int kk = 0; kk < 6; ++kk) {
        const float* p = &g2d[l16 * FV_STRIDE + 4 * kk + 2 * half];
        a[kk] = *(const v2f*)p;                    // ds_load_b64
      }

      // ---- Phase 1: issue ALL feat loads for this step up front, so their
      // latency hides behind the max/exp/WMMA chain and only one DScnt wait
      // is needed before the writeback pass.
      float f0[8], f1[8];
#pragma unroll
      for (int r = 0; r < 8; ++r) {
        const int b = r + 8 * half;
#if USE_TDM
        const float* fp = fb + ((size_t)b * CHUNK_T + tt) * S;
#else
        const float* fp = gchunk + ((size_t)b * TT + t) * S;
#endif
        f0[r] = fp[l16];        // N = 0..15
        f1[r] = fp[16 + l16];   // N = 16..23 (24..31 lands in padding)
      }

      // ---- Row max of g: 12 local values, VALU half swap. ----
      float m = fmaxf(a[0].x, a[0].y);
#pragma unroll
      for (int kk = 1; kk < 6; ++kk) m = fmaxf(m, fmaxf(a[kk].x, a[kk].y));
      m = fmaxf(m, halfswap16(m));

      // ---- P = exp2(g - m) in place: this IS the A operand. ----
#pragma unroll
      for (int kk = 0; kk < 6; ++kk) {
        a[kk].x = fast_exp2(a[kk].x - m);
        a[kk].y = fast_exp2(a[kk].y - m);
      }

      // ---- D = P x E : 12 chained WMMAs. ----
      v8f d0 = {};  // N = 0..15
      v8f d1 = {};  // N = 16..23
#pragma unroll
      for (int kk = 0; kk < 6; ++kk) {
        d0 = __builtin_amdgcn_wmma_f32_16x16x4_f32(
                 false, a[kk], false, bA[kk], (short)0, d0, false, false);
        d1 = __builtin_amdgcn_wmma_f32_16x16x4_f32(
                 false, a[kk], false, bB[kk], (short)0, d1, false, false);
      }

      // ---- Per-row offset accumulates rowmax while the row is alive. ----
      c += (t < lenL) ? m : 0.0f;

      // ---- Phase 2: g'[b][i] = log2(D) + feat/ln2, branch-free stores:
      // dead lanes / frozen rows store into the dump word instead.
#pragma unroll
      for (int r = 0; r < 8; ++r) {
        const int  b     = r + 8 * half;
        const bool alive = (t < len_r[r]);
        {
          const int i = l16;                       // N = 0..15
          const float v = __builtin_fmaf(f0[r], INV_LN2f, fast_log2(d0[r]));
          float* dst = alive ? &g2d[b * FV_STRIDE + i] : &dump[lane];
          *dst = v;
        }
        {
          const int i = 16 + l16;                  // N = 16..23 (24..31 pad)
          const float v = __builtin_fmaf(f1[r], INV_LN2f, fast_log2(d1[r]));
          float* dst = (alive && l16 < 8) ? &g2d[b * FV_STRIDE + i]
                                          : &dump[lane];
          *dst = v;
        }
      }
    }
#if USE_TDM
    buf ^= 1;
#endif
  }

  // ---- alpha[b] = ln2 * log2sumexp2_i(g[b][i] + c_b + trans[STOP][i]/ln2) ----
  const float tstop2 =
      (lane < S) ? trans[STOP_TAG * S + lane] * INV_LN2f : 0.0f;
  for (int b = 0; b < 16; ++b) {
    const float cb = __shfl(c, b, 32);
    const float v = (lane < S) ? g2d[b * FV_STRIDE + lane] + cb + tstop2
                               : -__builtin_inff();
    float mx = v;
#pragma unroll
    for (int off = 16; off > 0; off >>= 1) mx = fmaxf(mx, __shfl_xor(mx, off, 32));
    float sm = (lane < S) ? fast_exp2(v - mx) : 0.0f;
#pragma unroll
    for (int off = 16; off > 0; off >>= 1) sm += __shfl_xor(sm, off, 32);
    if (lane == 0) alpha_out[b0 + b] = (fast_log2(sm) + mx) * LN2f;
  }
}

// ---------------------------------------------------------------------------
// Gold path score: one wave per batch row, lanes stride over t.
// ---------------------------------------------------------------------------
__global__ __launch_bounds__(32) void crf_gold(
    const float* __restrict__ feats,
    const float* __restrict__ trans,
    const int*   __restrict__ tags,
    const int*   __restrict__ lengths,
    float*       __restrict__ gold_out)
{
  const int b    = blockIdx.x;
  const int lane = threadIdx.x;
  const int len  = lengths[b];
  const int*   tg = tags  + (size_t)b * TT;
  const float* ft = feats + (size_t)b * TT * S;

  float acc = 0.0f;
  for (int t = lane; t < len; t += 32) {
    const int cur  = tg[t];
    const int prev = (t == 0) ? START_TAG : tg[t - 1];
    acc += trans[cur * S + prev] + ft[(size_t)t * S + cur];
  }
#pragma unroll
  for (int off = 16; off > 0; off >>= 1) acc += __shfl_xor(acc, off, 32);

  if (lane == 0) {
    const int last = tg[len - 1];
    gold_out[b] = acc + trans[STOP_TAG * S + last];
  }
}

// ---------------------------------------------------------------------------
// Final reduction: mean(alpha - gold) -> out[0]
// ---------------------------------------------------------------------------
__global__ __launch_bounds__(256) void crf_finish(
    const float* __restrict__ alpha,
    const float* __restrict__ gold,
    float*       __restrict__ out)
{
  __shared__ float red[256];
  const int tid = threadIdx.x;
  float s = 0.0f;
  for (int b = tid; b < BB; b += 256) s += alpha[b] - gold[b];
  red[tid] = s;
  __syncthreads();
  for (int k = 128; k > 0; k >>= 1) {
    if (tid < k) red[tid] += red[tid + k];
    __syncthreads();
  }
  if (tid == 0) out[0] = red[0] * (1.0f / BB);
}

// ---------------------------------------------------------------------------
extern "C" void kernel_launch(void* const* d_in, const int* in_sizes, int n_in,
                              void* d_out, int out_size, void* d_ws, size_t ws_size,
                              hipStream_t stream) {
  const float* feats   = (const float*)d_in[0];
  const float* trans   = (const float*)d_in[1];
  const int*   tags    = (const int*)  d_in[2];
  const int*   lengths = (const int*)  d_in[3];

  float* alpha = (float*)d_ws;
  float* gold  = alpha + BB;

  crf_forward_wmma<<<BB / 16, 32, 0, stream>>>(feats, trans, lengths, alpha);
  crf_gold<<<BB, 32, 0, stream>>>(feats, trans, tags, lengths, gold);
  crf_finish<<<1, 256, 0, stream>>>(alpha, gold, (float*)d_out);
}